// ProteinLandscapeAnalyzer_33277406609662
// MI455X (gfx1250) — compile-verified
//
#include <hip/hip_runtime.h>
#include <math.h>

#define BB 4
#define NN 4096
#define DD 64
#define HH 128
#define H2 64
#define NSTATES 6
#define EPSF 1e-12f

// padded LDS strides to dodge 64-bank conflicts
#define H1S 132
#define H2S 68
#define DTS 17

typedef float v2f __attribute__((ext_vector_type(2)));
typedef float v8f __attribute__((ext_vector_type(8)));

__device__ __forceinline__ v8f wmma4(v2f a, v2f b, v8f c) {
  // V_WMMA_F32_16X16X4_F32 : D(16x16,f32) = A(16x4,f32) x B(4x16,f32) + C
  return __builtin_amdgcn_wmma_f32_16x16x4_f32(
      /*neg_a=*/false, a, /*neg_b=*/false, b,
      /*c_mod=*/(short)0, c, /*reuse_a=*/false, /*reuse_b=*/false);
}

// ---------------------------------------------------------------------------
// Kernel 1: energy MLP (e), state logits, plus per-row sq(|conf|^2) and nd.
// One wave32 per 16-row tile.
// ---------------------------------------------------------------------------
__global__ __launch_bounds__(32) void k_mlp(
    const float* __restrict__ conf, const float* __restrict__ native,
    const float* __restrict__ We1, const float* __restrict__ be1,
    const float* __restrict__ We2, const float* __restrict__ be2,
    const float* __restrict__ We3, const float* __restrict__ be3,
    const float* __restrict__ Ws1, const float* __restrict__ bs1,
    const float* __restrict__ Ws2, const float* __restrict__ bs2,
    float* __restrict__ out_e, float* __restrict__ out_logits,
    float* __restrict__ ws_sq, float* __restrict__ ws_nd)
{
  __shared__ float lh1[16 * H1S];
  __shared__ float lh2[16 * H2S];

  const int row0 = blockIdx.x * 16;          // global row among B*N
  const int b    = row0 / NN;
  const int lane = threadIdx.x;
  const bool lo  = lane < 16;
  const int m    = lo ? lane : lane - 16;    // row (A) / col (B,C) index
  const int kof  = lo ? 0 : 2;               // k sub-offset per lane half
  const int rb   = lo ? 0 : 8;               // C-layout row base

  const float* crow = conf + (size_t)(row0 + m) * DD;

  // A-tile: conf rows in WMMA f32 A layout, 16 K-steps of 4
  v2f a[16];
#pragma unroll
  for (int s = 0; s < 16; ++s) a[s] = *(const v2f*)(crow + 4 * s + kof);

  // ---- layer 1 (energy): h1 = relu(conf @ We1 + be1), 8 col-tiles of 16
#pragma unroll 1
  for (int t = 0; t < 8; ++t) {
    const int col = t * 16 + m;
    v8f ac0 = {}, ac1 = {}, ac2 = {}, ac3 = {};
#pragma unroll
    for (int s = 0; s < 16; s += 4) {
      v2f b0, b1, b2, b3;
      b0.x = We1[(4 * (s + 0) + kof) * HH + col];
      b0.y = We1[(4 * (s + 0) + kof + 1) * HH + col];
      b1.x = We1[(4 * (s + 1) + kof) * HH + col];
      b1.y = We1[(4 * (s + 1) + kof + 1) * HH + col];
      b2.x = We1[(4 * (s + 2) + kof) * HH + col];
      b2.y = We1[(4 * (s + 2) + kof + 1) * HH + col];
      b3.x = We1[(4 * (s + 3) + kof) * HH + col];
      b3.y = We1[(4 * (s + 3) + kof + 1) * HH + col];
      ac0 = wmma4(a[s + 0], b0, ac0);
      ac1 = wmma4(a[s + 1], b1, ac1);
      ac2 = wmma4(a[s + 2], b2, ac2);
      ac3 = wmma4(a[s + 3], b3, ac3);
    }
    v8f acc = (ac0 + ac1) + (ac2 + ac3);
    const float bias = be1[col];
#pragma unroll
    for (int r = 0; r < 8; ++r) {
      float v = acc[r] + bias;
      lh1[(rb + r) * H1S + col] = v > 0.f ? v : 0.f;
    }
  }
  __syncthreads();

  // ---- layer 2: h2 = relu(h1 @ We2 + be2), K=128 -> 32 steps
#pragma unroll 1
  for (int t = 0; t < 4; ++t) {
    const int col = t * 16 + m;
    v8f ac0 = {}, ac1 = {}, ac2 = {}, ac3 = {};
#pragma unroll
    for (int s = 0; s < 32; s += 4) {
#pragma unroll
      for (int q = 0; q < 4; ++q) {
        v2f av, bv;
        av.x = lh1[m * H1S + 4 * (s + q) + kof];
        av.y = lh1[m * H1S + 4 * (s + q) + kof + 1];
        bv.x = We2[(4 * (s + q) + kof) * H2 + col];
        bv.y = We2[(4 * (s + q) + kof + 1) * H2 + col];
        if (q == 0)      ac0 = wmma4(av, bv, ac0);
        else if (q == 1) ac1 = wmma4(av, bv, ac1);
        else if (q == 2) ac2 = wmma4(av, bv, ac2);
        else             ac3 = wmma4(av, bv, ac3);
      }
    }
    v8f acc = (ac0 + ac1) + (ac2 + ac3);
    const float bias = be2[col];
#pragma unroll
    for (int r = 0; r < 8; ++r) {
      float v = acc[r] + bias;
      lh2[(rb + r) * H2S + col] = v > 0.f ? v : 0.f;
    }
  }
  __syncthreads();

  // ---- layer 3 (scalar per row) + sq + nd
  if (lo) {
    float e = be3[0];
#pragma unroll
    for (int k = 0; k < H2; ++k) e += lh2[m * H2S + k] * We3[k];
    out_e[row0 + m] = e;

    const float* nat = native + (size_t)b * DD;
    float sq = 0.f, nd2 = 0.f;
#pragma unroll
    for (int k = 0; k < DD; ++k) {
      float c = crow[k];
      sq += c * c;
      float d = c - nat[k];
      nd2 += d * d;
    }
    ws_sq[row0 + m] = sq;
    ws_nd[row0 + m] = sqrtf(nd2);
  }
  __syncthreads();

  // ---- state path layer 1: h1s = relu(conf @ Ws1 + bs1)  (reuse lh1)
#pragma unroll 1
  for (int t = 0; t < 8; ++t) {
    const int col = t * 16 + m;
    v8f ac0 = {}, ac1 = {}, ac2 = {}, ac3 = {};
#pragma unroll
    for (int s = 0; s < 16; s += 4) {
#pragma unroll
      for (int q = 0; q < 4; ++q) {
        v2f bv;
        bv.x = Ws1[(4 * (s + q) + kof) * HH + col];
        bv.y = Ws1[(4 * (s + q) + kof + 1) * HH + col];
        if (q == 0)      ac0 = wmma4(a[s + q], bv, ac0);
        else if (q == 1) ac1 = wmma4(a[s + q], bv, ac1);
        else if (q == 2) ac2 = wmma4(a[s + q], bv, ac2);
        else             ac3 = wmma4(a[s + q], bv, ac3);
      }
    }
    v8f acc = (ac0 + ac1) + (ac2 + ac3);
    const float bias = bs1[col];
#pragma unroll
    for (int r = 0; r < 8; ++r) {
      float v = acc[r] + bias;
      lh1[(rb + r) * H1S + col] = v > 0.f ? v : 0.f;
    }
  }
  __syncthreads();

  // ---- logits = h1s @ Ws2 + bs2 (6 cols, scalar per row)
  if (lo) {
#pragma unroll
    for (int c = 0; c < NSTATES; ++c) {
      float acc = bs2[c];
      for (int k = 0; k < HH; ++k) acc += lh1[m * H1S + k] * Ws2[k * NSTATES + c];
      out_logits[(size_t)(row0 + m) * NSTATES + c] = acc;
    }
  }
}

// ---------------------------------------------------------------------------
// Kernel 2 (pass 1): per-row top-10 nearest distances via WMMA gram tiles.
// Produces r = 10th-smallest dist, is_basin from 9 nearest-neighbor energies.
// ---------------------------------------------------------------------------
__global__ __launch_bounds__(32) void k_topk(
    const float* __restrict__ conf, const float* __restrict__ e_glob,
    const float* __restrict__ ws_sq,
    float* __restrict__ out_basin, float* __restrict__ ws_r)
{
  __shared__ float ldist[16 * DTS];
  __shared__ float lej[16];
  __shared__ float lsqi[16];

  const int row0 = blockIdx.x * 16;
  const int b    = row0 / NN;
  const int lane = threadIdx.x;
  const bool lo  = lane < 16;
  const int m    = lo ? lane : lane - 16;
  const int kof  = lo ? 0 : 2;
  const int rb   = lo ? 0 : 8;

  const float* arow = conf + (size_t)(row0 + m) * DD;
  v2f a[16];
#pragma unroll
  for (int s = 0; s < 16; ++s) a[s] = *(const v2f*)(arow + 4 * s + kof);

  if (lo) lsqi[m] = ws_sq[row0 + m];
  __syncthreads();
  float sqrow[8];
#pragma unroll
  for (int r = 0; r < 8; ++r) sqrow[r] = lsqi[rb + r];

  float d10[10], e10[10];
#pragma unroll
  for (int k = 0; k < 10; ++k) { d10[k] = 3.0e38f; e10[k] = 0.f; }

  const float* cb  = conf  + (size_t)b * NN * DD;
  const float* eb  = e_glob + (size_t)b * NN;
  const float* sqb = ws_sq + (size_t)b * NN;

  for (int j0 = 0; j0 < NN; j0 += 16) {
    const float* brow = cb + (size_t)(j0 + m) * DD;
    __builtin_prefetch(brow + 16 * DD, 0, 0);   // global_prefetch next B tile
    v8f ac0 = {}, ac1 = {}, ac2 = {}, ac3 = {};
#pragma unroll
    for (int s = 0; s < 16; s += 4) {
      v2f b0 = *(const v2f*)(brow + 4 * (s + 0) + kof);
      v2f b1 = *(const v2f*)(brow + 4 * (s + 1) + kof);
      v2f b2 = *(const v2f*)(brow + 4 * (s + 2) + kof);
      v2f b3 = *(const v2f*)(brow + 4 * (s + 3) + kof);
      ac0 = wmma4(a[s + 0], b0, ac0);
      ac1 = wmma4(a[s + 1], b1, ac1);
      ac2 = wmma4(a[s + 2], b2, ac2);
      ac3 = wmma4(a[s + 3], b3, ac3);
    }
    v8f acc = (ac0 + ac1) + (ac2 + ac3);
    const float sqj = sqb[j0 + m];
    __syncthreads();
#pragma unroll
    for (int r = 0; r < 8; ++r) {
      float d2 = sqrow[r] + sqj - 2.f * acc[r];
      d2 = d2 < EPSF ? EPSF : d2;
      ldist[(rb + r) * DTS + m] = sqrtf(d2);
    }
    if (lo) lej[m] = eb[j0 + m];
    __syncthreads();
    if (lo) {
#pragma unroll 1
      for (int n = 0; n < 16; ++n) {
        float d = ldist[m * DTS + n];
        if (d < d10[9]) {                 // rare after warm-up
          float ej = lej[n];
#pragma unroll
          for (int k = 9; k >= 0; --k) {  // stable compare/shift insert (static idx)
            float dk   = d10[k];
            float dkm1 = (k > 0) ? d10[k - 1] : -1.f;
            float ekm1 = (k > 0) ? e10[k - 1] : 0.f;
            bool ltk   = d < dk;
            bool ltkm1 = (k > 0) && (d < dkm1);
            d10[k] = ltk ? (ltkm1 ? dkm1 : d) : dk;
            e10[k] = ltk ? (ltkm1 ? ekm1 : ej) : e10[k];
          }
        }
      }
    }
    __syncthreads();
  }

  if (lo) {
    ws_r[row0 + m] = d10[9];              // K-th smallest distance
    float ei = e_glob[row0 + m];
    float emin = e10[1];                  // skip self (index 0)
#pragma unroll
    for (int k = 2; k < 10; ++k) emin = fminf(emin, e10[k]);
    out_basin[row0 + m] = (ei <= emin) ? 1.0f : 0.0f;
  }
}

// ---------------------------------------------------------------------------
// Kernel 3 (pass 2): recompute dist tiles, masked depth/width with known r.
// ---------------------------------------------------------------------------
__global__ __launch_bounds__(32) void k_basin(
    const float* __restrict__ conf, const float* __restrict__ e_glob,
    const float* __restrict__ ws_sq, const float* __restrict__ ws_r,
    float* __restrict__ out_depth, float* __restrict__ out_width)
{
  __shared__ float ldist[16 * DTS];
  __shared__ float lej[16];
  __shared__ float lsqi[16];

  const int row0 = blockIdx.x * 16;
  const int b    = row0 / NN;
  const int lane = threadIdx.x;
  const bool lo  = lane < 16;
  const int m    = lo ? lane : lane - 16;
  const int kof  = lo ? 0 : 2;
  const int rb   = lo ? 0 : 8;

  const float* arow = conf + (size_t)(row0 + m) * DD;
  v2f a[16];
#pragma unroll
  for (int s = 0; s < 16; ++s) a[s] = *(const v2f*)(arow + 4 * s + kof);

  if (lo) lsqi[m] = ws_sq[row0 + m];
  __syncthreads();
  float sqrow[8];
#pragma unroll
  for (int r = 0; r < 8; ++r) sqrow[r] = lsqi[rb + r];

  float ri = 0.f, ei = 0.f;
  if (lo) { ri = ws_r[row0 + m]; ei = e_glob[row0 + m]; }
  float maxe = -3.0e38f, wsum = 0.f, cnt = 0.f;

  const float* cb  = conf  + (size_t)b * NN * DD;
  const float* eb  = e_glob + (size_t)b * NN;
  const float* sqb = ws_sq + (size_t)b * NN;

  for (int j0 = 0; j0 < NN; j0 += 16) {
    const float* brow = cb + (size_t)(j0 + m) * DD;
    __builtin_prefetch(brow + 16 * DD, 0, 0);
    v8f ac0 = {}, ac1 = {}, ac2 = {}, ac3 = {};
#pragma unroll
    for (int s = 0; s < 16; s += 4) {
      v2f b0 = *(const v2f*)(brow + 4 * (s + 0) + kof);
      v2f b1 = *(const v2f*)(brow + 4 * (s + 1) + kof);
      v2f b2 = *(const v2f*)(brow + 4 * (s + 2) + kof);
      v2f b3 = *(const v2f*)(brow + 4 * (s + 3) + kof);
      ac0 = wmma4(a[s + 0], b0, ac0);
      ac1 = wmma4(a[s + 1], b1, ac1);
      ac2 = wmma4(a[s + 2], b2, ac2);
      ac3 = wmma4(a[s + 3], b3, ac3);
    }
    v8f acc = (ac0 + ac1) + (ac2 + ac3);
    const float sqj = sqb[j0 + m];
    __syncthreads();
#pragma unroll
    for (int r = 0; r < 8; ++r) {
      float d2 = sqrow[r] + sqj - 2.f * acc[r];
      d2 = d2 < EPSF ? EPSF : d2;
      ldist[(rb + r) * DTS + m] = sqrtf(d2);
    }
    if (lo) lej[m] = eb[j0 + m];
    __syncthreads();
    if (lo) {
#pragma unroll
      for (int n = 0; n < 16; ++n) {
        float d = ldist[m * DTS + n];
        if (d < ri) {
          maxe = fmaxf(maxe, lej[n]);
          wsum += d;
          cnt  += 1.f;
        }
      }
    }
    __syncthreads();
  }

  if (lo) {
    out_depth[row0 + m] = maxe - ei;   // mask always contains self -> maxe finite
    out_width[row0 + m] = wsum / cnt;
  }
}

// ---------------------------------------------------------------------------
// Kernel 4: per-batch metrics. One 256-thread block per batch.
// ---------------------------------------------------------------------------
__global__ __launch_bounds__(256) void k_metrics(
    const float* __restrict__ e_glob, const float* __restrict__ ws_nd,
    float* __restrict__ out_metrics)
{
  __shared__ double dred[256];
  __shared__ float  fred[256];
  __shared__ int    ired[256];
  __shared__ float  lse[NN];           // e sorted by nd-rank

  const int b   = blockIdx.x;
  const int tid = threadIdx.x;
  const float* e  = e_glob + (size_t)b * NN;
  const float* nd = ws_nd  + (size_t)b * NN;

  double se = 0, se2 = 0, snd = 0, snd2 = 0, snde = 0;
  float mx = -3.0e38f, mn = 3.0e38f;
  int mni = 0;
  for (int i = tid; i < NN; i += 256) {
    float ev = e[i], nv = nd[i];
    se += ev; se2 += (double)ev * ev;
    snd += nv; snd2 += (double)nv * nv;
    snde += (double)nv * ev;
    mx = fmaxf(mx, ev);
    if (nv < mn) { mn = nv; mni = i; }
  }

  // 5 double-sum reductions
  double part[5] = { se, se2, snd, snd2, snde };
  double tot[5];
  for (int q = 0; q < 5; ++q) {
    dred[tid] = part[q];
    __syncthreads();
    for (int off = 128; off > 0; off >>= 1) {
      if (tid < off) dred[tid] += dred[tid + off];
      __syncthreads();
    }
    tot[q] = dred[0];
    __syncthreads();
  }

  // argmin(nd) with first-index tie rule
  fred[tid] = mn; ired[tid] = mni;
  __syncthreads();
  for (int off = 128; off > 0; off >>= 1) {
    if (tid < off) {
      float ov = fred[tid + off]; int oi = ired[tid + off];
      if (ov < fred[tid] || (ov == fred[tid] && oi < ired[tid])) {
        fred[tid] = ov; ired[tid] = oi;
      }
    }
    __syncthreads();
  }
  const int amin = ired[0];
  __syncthreads();

  // max(e)
  fred[tid] = mx;
  __syncthreads();
  for (int off = 128; off > 0; off >>= 1) {
    if (tid < off) fred[tid] = fmaxf(fred[tid], fred[tid + off]);
    __syncthreads();
  }
  const float maxE = fred[0];
  __syncthreads();

  // stable rank of nd -> scatter e into lse (rank is a permutation)
  for (int i = tid; i < NN; i += 256) {
    float ndi = nd[i];
    int cnt = 0;
    for (int j = 0; j < NN; ++j) {
      float ndj = nd[j];
      cnt += (ndj < ndi) || (ndj == ndi && j < i);
    }
    lse[cnt] = e[i];
  }
  __syncthreads();

  int fc = 0;
  for (int k = tid + 1; k < NN; k += 256) fc += (lse[k] > lse[k - 1]) ? 1 : 0;
  ired[tid] = fc;
  __syncthreads();
  for (int off = 128; off > 0; off >>= 1) {
    if (tid < off) ired[tid] += ired[tid + off];
    __syncthreads();
  }

  if (tid == 0) {
    const double n = (double)NN;
    double me  = tot[0] / n;
    double mnd = tot[2] / n;
    double ve  = tot[1] - n * me * me;
    double vnd = tot[3] - n * mnd * mnd;
    double cov = tot[4] - n * mnd * me;
    double ruggedness = sqrt(fmax(ve / (n - 1.0), 0.0));
    double corr = cov / sqrt(vnd * ve);
    float native_e = e[amin];
    float funnel = maxE - native_e;
    float frustration = (float)ired[0] / (float)(NN - 1);
    out_metrics[b * 5 + 0] = funnel;
    out_metrics[b * 5 + 1] = (float)ruggedness;
    out_metrics[b * 5 + 2] = (float)corr;
    out_metrics[b * 5 + 3] = frustration;
    out_metrics[b * 5 + 4] = native_e;
  }
}

// ---------------------------------------------------------------------------
extern "C" void kernel_launch(void* const* d_in, const int* in_sizes, int n_in,
                              void* d_out, int out_size, void* d_ws, size_t ws_size,
                              hipStream_t stream) {
  const float* conf   = (const float*)d_in[0];
  const float* native = (const float*)d_in[1];
  const float* We1 = (const float*)d_in[2];
  const float* be1 = (const float*)d_in[3];
  const float* We2 = (const float*)d_in[4];
  const float* be2 = (const float*)d_in[5];
  const float* We3 = (const float*)d_in[6];
  const float* be3 = (const float*)d_in[7];
  const float* Ws1 = (const float*)d_in[8];
  const float* bs1 = (const float*)d_in[9];
  const float* Ws2 = (const float*)d_in[10];
  const float* bs2 = (const float*)d_in[11];

  float* out = (float*)d_out;
  const size_t BN = (size_t)BB * NN;
  float* out_e       = out;
  float* out_basin   = out + BN;
  float* out_depth   = out + 2 * BN;
  float* out_width   = out + 3 * BN;
  float* out_logits  = out + 4 * BN;
  float* out_metrics = out + 10 * BN;

  float* ws   = (float*)d_ws;
  float* ws_sq = ws;
  float* ws_nd = ws + BN;
  float* ws_r  = ws + 2 * BN;

  const int ntiles = (BB * NN) / 16;   // 1024 wave32 tiles

  k_mlp<<<ntiles, 32, 0, stream>>>(conf, native, We1, be1, We2, be2, We3, be3,
                                   Ws1, bs1, Ws2, bs2,
                                   out_e, out_logits, ws_sq, ws_nd);
  k_topk<<<ntiles, 32, 0, stream>>>(conf, out_e, ws_sq, out_basin, ws_r);
  k_basin<<<ntiles, 32, 0, stream>>>(conf, out_e, ws_sq, ws_r, out_depth, out_width);
  k_metrics<<<BB, 256, 0, stream>>>(out_e, ws_nd, out_metrics);
}